// SSIM_45140106281601
// MI455X (gfx1250) — compile-verified
//
#include <hip/hip_runtime.h>

typedef __attribute__((ext_vector_type(2))) float v2f;
typedef __attribute__((ext_vector_type(8))) float v8f;

// ---------------- problem geometry ----------------
#define IMG_H   512
#define IMG_W   512
#define WIN     11
#define OUT_H   502
#define OUT_W   502
#define NPLANES 48                 // B*C = 16*3
#define PLANE_STRIDE (IMG_H * IMG_W)

// ---------------- tiling ----------------
#define TOX 64                     // output tile width
#define TOY 64                     // output tile height
#define INROWS 80                  // 5 chunks of 16 rows (covers 64+10 halo + K=28 reach)
#define INCOLS 76                  // 64 + 10 halo + 2 (K=28 reach)
#define SSTR   78                  // EVEN input-tile stride -> 8B-aligned v2f LDS loads,
                                   // 14n mod 64 distinct -> conflict-free A reads
#define HSTR   82                  // EVEN transposed-H stride -> aligned b64 loads/stores

#define NTHREADS 256
#define NWAVES   8

#define C1 1.0e-4f                 // (0.01*1.0)^2
#define C2 9.0e-4f                 // (0.03*1.0)^2
#define INV_N (1.0f / 12096192.0f) // 1 / (48*502*502)

__device__ __forceinline__ int imin(int a, int b) { return a < b ? a : b; }

// Banded Gaussian weight lookup: returns w[d] for d in [0,10], else 0.
// sw lives in LDS (per-lane dynamic index -> ds_load, no scratch spill).
__device__ __forceinline__ float wtap(const float* sw, int d) {
    const int dc = d < 0 ? 0 : (d > 10 ? 10 : d);
    const float v = sw[dc];
    return (d == dc) ? v : 0.f;
}

#define WMMA(Av, Bv, Cv) \
    __builtin_amdgcn_wmma_f32_16x16x4_f32(false, (Av), false, (Bv), (short)0, (Cv), false, false)

// ---------------------------------------------------------------------------
// Fused SSIM: async tiles -> LDS, H-blur of 5 fields as banded GEMM on
// v_wmma_f32_16x16x4_f32, transposed H in LDS, V-blur as banded GEMM,
// elementwise SSIM on the aligned accumulators, wave reduce, one atomic.
// ---------------------------------------------------------------------------
__global__ __launch_bounds__(NTHREADS)
void ssim_wmma_kernel(const float* __restrict__ A,
                      const float* __restrict__ B,
                      const float* __restrict__ kern,
                      float* __restrict__ out) {
    __shared__ float sA [INROWS * SSTR];   // 80x78 input tile (a)
    __shared__ float sB [INROWS * SSTR];   // 80x78 input tile (b)
    __shared__ float HTa [TOX * HSTR];     // h-blurred fields, TRANSPOSED [col][row]
    __shared__ float HTb [TOX * HSTR];
    __shared__ float HTaa[TOX * HSTR];
    __shared__ float HTbb[TOX * HSTR];
    __shared__ float HTab[TOX * HSTR];
    __shared__ float sw[16];
    __shared__ float red[NWAVES];

    const int tid   = threadIdx.x;
    const int lane  = tid & 31;
    // Wave id as an SGPR: job loops become scalar-uniform (no exec-mask
    // bookkeeping), and EXEC is provably all-1s around every WMMA.
    const int wv    = __builtin_amdgcn_readfirstlane(tid >> 5);
    const int h     = (lane >> 4) & 1;     // lane half (WMMA K split)
    const int n     = lane & 15;           // M (A rows) / N (B cols) index
    const int plane = blockIdx.z;
    const int gx0   = blockIdx.x * TOX;
    const int gy0   = blockIdx.y * TOY;

    if (tid < WIN) sw[tid] = kern[tid];

    // ------------------------------------------------------------------
    // Stage 1: async global -> LDS tile loads (GVS: lds vdst, v-offset, s-base)
    // ------------------------------------------------------------------
    const unsigned long long baseA =
        (unsigned long long)(A + (size_t)plane * PLANE_STRIDE);
    const unsigned long long baseB =
        (unsigned long long)(B + (size_t)plane * PLANE_STRIDE);

    for (int idx = tid; idx < INROWS * INCOLS; idx += NTHREADS) {
        const int r  = idx / INCOLS;
        const int c  = idx - r * INCOLS;
        const int gr = imin(gy0 + r, IMG_H - 1);   // clamped halo: finite values,
        const int gc = imin(gx0 + c, IMG_W - 1);   // only feed discarded outputs
        const int voff = (gr * IMG_W + gc) * 4;
        const unsigned la = (unsigned)(size_t)&sA[r * SSTR + c];
        const unsigned lb = (unsigned)(size_t)&sB[r * SSTR + c];
        asm volatile("global_load_async_to_lds_b32 %0, %1, %2"
                     :: "v"(la), "v"(voff), "s"(baseA) : "memory");
        asm volatile("global_load_async_to_lds_b32 %0, %1, %2"
                     :: "v"(lb), "v"(voff), "s"(baseB) : "memory");
    }
    asm volatile("s_wait_asynccnt 0" ::: "memory");
    __syncthreads();

    // ------------------------------------------------------------------
    // Per-lane banded weight fragment. Serves as B-operand of the H-pass
    // (Wh[k][n] = w[k-n]) AND A-operand of the V-pass (Wv[m][k] = w[k-m]):
    // both reduce to w[(4s+2h(+1)) - (lane&15)].
    // ------------------------------------------------------------------
    v2f wop[7];
#pragma unroll
    for (int s = 0; s < 7; ++s) {
        const int u = 4 * s + 2 * h;
        wop[s].x = wtap(sw, u - n);
        wop[s].y = wtap(sw, u + 1 - n);
    }

    // ------------------------------------------------------------------
    // Stage 2 (H-pass): H(16x16) = In(16x28) x Wh(28x16) per tile,
    // 5 fields share each A fragment (a, b, a*a, b*b, a*b).
    // 20 tiles (5 row-chunks x 4 col-chunks) round-robined over 8 waves.
    // ------------------------------------------------------------------
    for (int job = wv; job < 20; job += NWAVES) {
        const int r0 = (job >> 2) * 16;
        const int j0 = (job & 3) * 16;
        v8f cA = {}, cB = {}, cAA = {}, cBB = {}, cAB = {};
#pragma unroll
        for (int s = 0; s < 7; ++s) {
            const int off = (r0 + n) * SSTR + j0 + 4 * s + 2 * h;  // even -> b64
            const v2f va = *(const v2f*)&sA[off];
            const v2f vb = *(const v2f*)&sB[off];
            cA  = WMMA(va,      wop[s], cA);
            cB  = WMMA(vb,      wop[s], cB);
            cAA = WMMA(va * va, wop[s], cAA);
            cBB = WMMA(vb * vb, wop[s], cBB);
            cAB = WMMA(va * vb, wop[s], cAB);
        }
        // D layout: vgpr p <-> row r0+p+8h, col j0+n. Store transposed so the
        // V-pass B-operand becomes one aligned b64 per field.
        const int base = (j0 + n) * HSTR + r0 + 8 * h;             // even
#pragma unroll
        for (int p = 0; p < 8; p += 2) {
            v2f t;
            t.x = cA[p];  t.y = cA[p + 1];  *(v2f*)&HTa [base + p] = t;
            t.x = cB[p];  t.y = cB[p + 1];  *(v2f*)&HTb [base + p] = t;
            t.x = cAA[p]; t.y = cAA[p + 1]; *(v2f*)&HTaa[base + p] = t;
            t.x = cBB[p]; t.y = cBB[p + 1]; *(v2f*)&HTbb[base + p] = t;
            t.x = cAB[p]; t.y = cAB[p + 1]; *(v2f*)&HTab[base + p] = t;
        }
    }
    __syncthreads();

    // ------------------------------------------------------------------
    // Stage 3 (V-pass + SSIM): Out(16x16) = Wv(16x28) x H(28x16) per tile.
    // B-operand b = { H[k][n], H[k+1][n] } = consecutive floats in HT.
    // 16 tiles over 8 waves (exactly 2 each). Accumulators are
    // component-aligned -> SSIM is elementwise per lane (8 outputs each),
    // with the two quotients fused into a single v_rcp_f32.
    // ------------------------------------------------------------------
    float acc = 0.f;
    for (int job = wv; job < 16; job += NWAVES) {
        const int i0 = (job >> 2) * 16;
        const int j0 = (job & 3) * 16;
        v8f mA = {}, mB = {}, vAA = {}, vBB = {}, vAB = {};
#pragma unroll
        for (int s = 0; s < 7; ++s) {
            const int hb = (j0 + n) * HSTR + i0 + 4 * s + 2 * h;   // even -> b64
            mA  = WMMA(wop[s], *(const v2f*)&HTa [hb], mA);
            mB  = WMMA(wop[s], *(const v2f*)&HTb [hb], mB);
            vAA = WMMA(wop[s], *(const v2f*)&HTaa[hb], vAA);
            vBB = WMMA(wop[s], *(const v2f*)&HTbb[hb], vBB);
            vAB = WMMA(wop[s], *(const v2f*)&HTab[hb], vAB);
        }
#pragma unroll
        for (int p = 0; p < 8; ++p) {
            const int oy = gy0 + i0 + p + 8 * h;
            const int ox = gx0 + j0 + n;
            const float ma  = mA[p],  mb  = mB[p];
            const float ma2 = ma * ma, mb2 = mb * mb, mab = ma * mb;
            const float va  = vAA[p] - ma2;
            const float vb  = vBB[p] - mb2;
            const float vab = vAB[p] - mab;
            const float num = (2.f * vab + C2) * (2.f * mab + C1);
            const float den = (va + vb + C2) * (ma2 + mb2 + C1);
            const float ss  = num * __builtin_amdgcn_rcpf(den);
            acc += (oy < OUT_H && ox < OUT_W) ? ss : 0.f;
        }
    }

    // ------------------------------------------------------------------
    // Stage 4: wave shuffle reduction, one LDS word per wave, one atomic.
    // ------------------------------------------------------------------
#pragma unroll
    for (int m = 16; m > 0; m >>= 1) acc += __shfl_xor(acc, m, 32);
    if (lane == 0) red[wv] = acc;
    __syncthreads();
    if (tid == 0) {
        float s = 0.f;
#pragma unroll
        for (int i = 0; i < NWAVES; ++i) s += red[i];
        atomicAdd(out, s * INV_N);
    }
}

// ---------------------------------------------------------------------------
// Host launcher
// ---------------------------------------------------------------------------
extern "C" void kernel_launch(void* const* d_in, const int* in_sizes, int n_in,
                              void* d_out, int out_size, void* d_ws, size_t ws_size,
                              hipStream_t stream) {
    const float* a = (const float*)d_in[0];   // inputs_a [16,3,512,512] f32
    const float* b = (const float*)d_in[1];   // inputs_b [16,3,512,512] f32
    const float* k = (const float*)d_in[2];   // kernel   [11] f32
    float* out = (float*)d_out;               // scalar f32

    hipMemsetAsync(out, 0, sizeof(float), stream);   // graph-capture safe

    dim3 grid((OUT_W + TOX - 1) / TOX,   // 8
              (OUT_H + TOY - 1) / TOY,   // 8
              NPLANES);                  // 48
    ssim_wmma_kernel<<<grid, NTHREADS, 0, stream>>>(a, b, k, out);
}